// ANR_1700807049369
// MI455X (gfx1250) — compile-verified
//
#include <hip/hip_runtime.h>

typedef float v2f __attribute__((ext_vector_type(2)));
typedef float v8f __attribute__((ext_vector_type(8)));

#define VOCAB 50000
#define D     300
#define H     10
#define WW    3
#define AFF   50
#define A     5
#define NB    256          // batch
#define L     500
#define NWIN  498          // L - WW + 1
#define WH    30           // WW*H
#define NCOL  50           // A*H
#define NCOLP 64           // padded to 4 WMMA N-tiles
#define KPAD  304          // weight K stride (floats), D=300 used
#define MAST  52           // Ma LDS row stride (floats)

// ---------------------------------------------------------------------------
// Kernel 0: repack W_user/W_item [A][D][H] -> transposed padded [2][64][KPAD]
// wt[s][col][k] = W[s][col/10][k][col%10], zero outside col<50, k<300
// ---------------------------------------------------------------------------
__global__ void repack_weights(const float* __restrict__ W_user,
                               const float* __restrict__ W_item,
                               float* __restrict__ wt) {
    int idx = blockIdx.x * blockDim.x + threadIdx.x;
    const int per = NCOLP * KPAD;
    if (idx >= 2 * per) return;
    int s = idx / per;
    int rem = idx - s * per;
    int col = rem / KPAD;
    int k = rem - col * KPAD;
    float v = 0.0f;
    if (col < NCOL && k < D) {
        int a = col / H, h = col - a * H;
        const float* src = (s == 0) ? W_user : W_item;
        v = src[(a * D + k) * H + h];
    }
    wt[idx] = v;
}

// ---------------------------------------------------------------------------
// Kernel 1: fused  gather -> WMMA projection (f32 16x16x4) -> attention pool
// one block per (stream, b); Ma kept entirely in LDS.
// Each wave owns a 16-row L-tile and computes all 4 N-tiles (A-frag reuse x4):
// per k-step: 1 A global_load_b64 + 4 B global_load_b64 + 4 v_wmma.
// No barriers in the projection -> full async pipelining across waves.
// ---------------------------------------------------------------------------
__global__ void __launch_bounds__(256, 1)
anr_pool(const float* __restrict__ word_emb,
         const int* __restrict__ user_review,
         const int* __restrict__ item_review,
         const float* __restrict__ aspect_vec,
         const float* __restrict__ wt,
         float* __restrict__ pq) {
    __shared__ float s_ma[L * MAST];       // 104 KB : projected Ma[b] (500 x 50)
    __shared__ float s_att[NWIN];          // attention weights
    __shared__ float s_red[256];           // reductions

    const int b    = blockIdx.x;
    const int strm = blockIdx.y;           // 0 = user, 1 = item
    const int tid  = threadIdx.x;
    const int lane = tid & 31;
    const int wave = tid >> 5;
    const int lane15 = lane & 15;

    const int* review = (strm == 0 ? user_review : item_review) + b * L;
    const float* wts  = wt + (size_t)strm * NCOLP * KPAD;

    // WMMA f32 16x16x4 fragment addressing (ISA 7.12.2):
    // lane<16 holds K {4k,4k+1}, lane>=16 holds K {4k+2,4k+3}
    const int koff = (lane < 16) ? 0 : 2;
    const float* bp0 = wts + (size_t)(0 * 16 + lane15) * KPAD + koff;
    const float* bp1 = wts + (size_t)(1 * 16 + lane15) * KPAD + koff;
    const float* bp2 = wts + (size_t)(2 * 16 + lane15) * KPAD + koff;
    const float* bp3 = wts + (size_t)(3 * 16 + lane15) * KPAD + koff;

    for (int t = wave; t < 32; t += 8) {          // 32 L-tiles of 16 rows
        int l = t * 16 + lane15;
        if (l > L - 1) l = L - 1;                 // pad rows: clamp (never stored)
        int tok = review[l];
        const float* ap = word_emb + (size_t)tok * D + koff;

        v8f c0 = {0.f,0.f,0.f,0.f,0.f,0.f,0.f,0.f};
        v8f c1 = c0, c2 = c0, c3 = c0;
        for (int ks = 0; ks < D / 4; ++ks) {
            v2f av = *(const v2f*)(ap  + ks * 4);
            v2f b0 = *(const v2f*)(bp0 + ks * 4);
            v2f b1 = *(const v2f*)(bp1 + ks * 4);
            v2f b2 = *(const v2f*)(bp2 + ks * 4);
            v2f b3 = *(const v2f*)(bp3 + ks * 4);
            c0 = __builtin_amdgcn_wmma_f32_16x16x4_f32(false, av, false, b0, (short)0, c0, false, false);
            c1 = __builtin_amdgcn_wmma_f32_16x16x4_f32(false, av, false, b1, (short)0, c1, false, false);
            c2 = __builtin_amdgcn_wmma_f32_16x16x4_f32(false, av, false, b2, (short)0, c2, false, false);
            c3 = __builtin_amdgcn_wmma_f32_16x16x4_f32(false, av, false, b3, (short)0, c3, false, false);
        }

        // C/D layout: VGPR r, lane l -> M = r + 8*(l>=16), N = l&15
        int rowbase = t * 16 + ((lane >= 16) ? 8 : 0);
        v8f cacc[4] = {c0, c1, c2, c3};
#pragma unroll
        for (int nt = 0; nt < 4; ++nt) {
            int col = nt * 16 + lane15;
            if (col < NCOL) {
#pragma unroll
                for (int r = 0; r < 8; ++r) {
                    int row = rowbase + r;
                    if (row < L) s_ma[row * MAST + col] = cacc[nt][r];
                }
            }
        }
    }
    __syncthreads();

    // =========================== attention pooling ==========================
    float* pqo = pq + ((size_t)strm * NB + b) * (A * WH);

    for (int a = 0; a < A; ++a) {
        float avr[WH];
#pragma unroll
        for (int w = 0; w < WH; ++w) avr[w] = aspect_vec[a * WH + w];

        // window scores s[n] = <z[n], aspect_vec[a]>
        float sc0 = -1e30f, sc1 = -1e30f;
        int n0 = tid, n1 = tid + 256;
        if (n0 < NWIN) {
            float s = 0.f;
#pragma unroll
            for (int j = 0; j < WW; ++j)
#pragma unroll
                for (int h = 0; h < H; ++h)
                    s += s_ma[(n0 + j) * MAST + a * H + h] * avr[j * H + h];
            sc0 = s;
        }
        if (n1 < NWIN) {
            float s = 0.f;
#pragma unroll
            for (int j = 0; j < WW; ++j)
#pragma unroll
                for (int h = 0; h < H; ++h)
                    s += s_ma[(n1 + j) * MAST + a * H + h] * avr[j * H + h];
            sc1 = s;
        }

        // block max
        s_red[tid] = fmaxf(sc0, sc1);
        __syncthreads();
        for (int s = 128; s > 0; s >>= 1) {
            if (tid < s) s_red[tid] = fmaxf(s_red[tid], s_red[tid + s]);
            __syncthreads();
        }
        float mx = s_red[0];
        __syncthreads();

        // exp + block sum; unnormalized att -> s_att
        float e0 = (n0 < NWIN) ? __expf(sc0 - mx) : 0.f;
        float e1 = (n1 < NWIN) ? __expf(sc1 - mx) : 0.f;
        if (n0 < NWIN) s_att[n0] = e0;
        if (n1 < NWIN) s_att[n1] = e1;
        s_red[tid] = e0 + e1;
        __syncthreads();
        for (int s = 128; s > 0; s >>= 1) {
            if (tid < s) s_red[tid] += s_red[tid + s];
            __syncthreads();
        }
        float denom = s_red[0];
        __syncthreads();

        // pooled[w] = (1/denom) * sum_n att[n] * Ma[n + w/H][a*H + w%H]
        if (tid < WH) {
            int j = tid / H, h = tid - j * H;
            float p = 0.f;
            for (int n = 0; n < NWIN; ++n)
                p += s_att[n] * s_ma[(n + j) * MAST + a * H + h];
            pqo[a * WH + tid] = p / denom;
        }
        __syncthreads();
    }
}

// ---------------------------------------------------------------------------
// Kernel 2: per-b co-attention combine (tiny)
// ---------------------------------------------------------------------------
__global__ void anr_final(const float* __restrict__ pq,
                          const float* __restrict__ W_s,
                          const float* __restrict__ W_x,
                          const float* __restrict__ W_y,
                          const float* __restrict__ v_x,
                          const float* __restrict__ v_y,
                          const float* __restrict__ b_users,
                          const float* __restrict__ b_items,
                          const float* __restrict__ b_0,
                          const int* __restrict__ uid,
                          const int* __restrict__ iid,
                          float* __restrict__ out) {
    __shared__ float P[A * WH], Q[A * WH], T[A * WH], S[A * A];
    __shared__ float PX[A * AFF], QY[A * AFF], HU[A * AFF], HI[A * AFF];
    __shared__ float bu[A], bi[A], pa[A];

    const int b = blockIdx.x, tid = threadIdx.x;
    const float* Pg = pq + (size_t)b * (A * WH);
    const float* Qg = pq + (size_t)(NB + b) * (A * WH);
    for (int i = tid; i < A * WH; i += 64) { P[i] = Pg[i]; Q[i] = Qg[i]; }
    __syncthreads();

    // T = P @ W_s
    for (int i = tid; i < A * WH; i += 64) {
        int a = i / WH, v = i - a * WH;
        float s = 0.f;
        for (int w = 0; w < WH; ++w) s += P[a * WH + w] * W_s[w * WH + v];
        T[i] = s;
    }
    __syncthreads();

    // S = relu(T @ Q^T);  PX = P@W_x; QY = Q@W_y
    for (int i = tid; i < A * A; i += 64) {
        int a = i / A, c2 = i - a * A;
        float s = 0.f;
        for (int v = 0; v < WH; ++v) s += T[a * WH + v] * Q[c2 * WH + v];
        S[i] = fmaxf(s, 0.f);
    }
    for (int i = tid; i < A * AFF; i += 64) {
        int a = i / AFF, f = i - a * AFF;
        float sx = 0.f, sy = 0.f;
        for (int w = 0; w < WH; ++w) {
            sx += P[a * WH + w] * W_x[w * AFF + f];
            sy += Q[a * WH + w] * W_y[w * AFF + f];
        }
        PX[i] = sx; QY[i] = sy;
    }
    __syncthreads();

    // H_u = relu(PX + S^T@QY); H_i = relu(QY + S@PX)
    for (int i = tid; i < A * AFF; i += 64) {
        int a = i / AFF, f = i - a * AFF;
        float su = PX[i], si = QY[i];
        for (int c2 = 0; c2 < A; ++c2) {
            su += S[c2 * A + a] * QY[c2 * AFF + f];
            si += S[a * A + c2] * PX[c2 * AFF + f];
        }
        HU[i] = fmaxf(su, 0.f);
        HI[i] = fmaxf(si, 0.f);
    }
    __syncthreads();

    if (tid < A) {
        float su = 0.f, si = 0.f, p = 0.f;
        for (int f = 0; f < AFF; ++f) {
            su += HU[tid * AFF + f] * v_x[f];
            si += HI[tid * AFF + f] * v_y[f];
        }
        for (int w = 0; w < WH; ++w) p += P[tid * WH + w] * Q[tid * WH + w];
        bu[tid] = su; bi[tid] = si; pa[tid] = p;
    }
    __syncthreads();

    if (tid == 0) {
        float mu = bu[0], mi = bi[0];
        for (int a2 = 1; a2 < A; ++a2) { mu = fmaxf(mu, bu[a2]); mi = fmaxf(mi, bi[a2]); }
        float su = 0.f, si = 0.f, eu[A], ei[A];
        for (int a2 = 0; a2 < A; ++a2) {
            eu[a2] = __expf(bu[a2] - mu); ei[a2] = __expf(bi[a2] - mi);
            su += eu[a2]; si += ei[a2];
        }
        float pred = 0.f;
        for (int a2 = 0; a2 < A; ++a2)
            pred += (eu[a2] / su) * (ei[a2] / si) * pa[a2];
        pred += b_users[uid[b]] + b_items[iid[b]] + b_0[0];
        out[b] = pred;
    }
}

// ---------------------------------------------------------------------------
extern "C" void kernel_launch(void* const* d_in, const int* in_sizes, int n_in,
                              void* d_out, int out_size, void* d_ws, size_t ws_size,
                              hipStream_t stream) {
    (void)in_sizes; (void)n_in; (void)out_size; (void)ws_size;
    const float* word_emb   = (const float*)d_in[0];
    const float* W_user     = (const float*)d_in[1];
    const float* W_item     = (const float*)d_in[2];
    const float* aspect_vec = (const float*)d_in[3];
    const float* W_s        = (const float*)d_in[4];
    const float* W_x        = (const float*)d_in[5];
    const float* W_y        = (const float*)d_in[6];
    const float* v_x        = (const float*)d_in[7];
    const float* v_y        = (const float*)d_in[8];
    const float* b_users    = (const float*)d_in[9];
    const float* b_items    = (const float*)d_in[10];
    const float* b_0        = (const float*)d_in[11];
    const int* user_review  = (const int*)d_in[12];
    const int* item_review  = (const int*)d_in[13];
    const int* uid          = (const int*)d_in[14];
    const int* iid          = (const int*)d_in[15];

    float* wt = (float*)d_ws;                                   // 2*64*304 f32
    float* pq = (float*)((char*)d_ws + (size_t)2 * NCOLP * KPAD * sizeof(float));

    repack_weights<<<(2 * NCOLP * KPAD + 255) / 256, 256, 0, stream>>>(W_user, W_item, wt);
    anr_pool<<<dim3(NB, 2), 256, 0, stream>>>(word_emb, user_review, item_review,
                                              aspect_vec, wt, pq);
    anr_final<<<NB, 64, 0, stream>>>(pq, W_s, W_x, W_y, v_x, v_y,
                                     b_users, b_items, b_0, uid, iid, (float*)d_out);
}